// SSIMLoss_38895223833172
// MI455X (gfx1250) — compile-verified
//
#include <hip/hip_runtime.h>

// ---------------------------------------------------------------------------
// SSIM loss, fused separable 11x11 Gaussian conv via V_WMMA_F32_16X16X4_F32.
// Layout facts used (CDNA5 ISA):
//   A 16x4 f32 : lane L holds row M=L%16; VGPR v holds K = v + 2*(L>=16)
//   B 4x16  f32: lane L holds col N=L%16; VGPR v holds row K = v + 2*(L>=16)
//   C/D 16x16  : lane L holds col N=L%16; VGPR v holds row M = v + 8*(L>=16)
// v3 changes vs round 2:
//   - per-wave H scratch is stored COLUMN-major (pitch 28): C/D->LDS becomes
//     4x ds_store_b64 per row block, vertical B-fragment pickup becomes one
//     aligned ds_load_b64 per chunk (ds ops ~214 -> ~130)
//   - pad rows 26/27 of each H column zeroed once (weight there is 0; zeroing
//     avoids NaN*0 from uninitialized LDS inside the b64 read)
// ---------------------------------------------------------------------------

typedef float v2f __attribute__((ext_vector_type(2)));
typedef float v8f __attribute__((ext_vector_type(8)));

#define SSIM_C1 0.0001f
#define SSIM_C2 0.0009f

#define IMG_H 512
#define IMG_W 512
#define PLANE (IMG_H * IMG_W)
#define NPLANES 96              // 32 * 3
#define TILES_X 8               // 64-wide workgroup tile
#define TILES_Y 16              // 32-tall workgroup tile
#define NWG (NPLANES * TILES_X * TILES_Y)   // 12288
#define NTOT 25165824.0f        // 32*3*512*512

#define LP 75                   // LDS pitch for input planes (odd -> no bank conflict)
#define HROWS 42                // 32 + 10 halo
#define HCOLS 74                // 64 + 10 halo
#define HPITCH 28               // per-wave H scratch: 16 columns x 28 rows (col-major)

// Normalized Gaussian taps, sigma=1.5, size 11
__device__ __constant__ float GW[11] = {
    0.00102840f, 0.00759876f, 0.03600074f, 0.10936099f, 0.21300553f,
    0.26601172f, 0.21300553f, 0.10936099f, 0.03600074f, 0.00759876f,
    0.00102840f};

static __device__ inline v8f wmma4(v2f a, v2f b, v8f c) {
    // D = A(16x4) * B(4x16) + C on the matrix pipe (fp32, exact-precision path)
    return __builtin_amdgcn_wmma_f32_16x16x4_f32(
        false, a, false, b, (short)0, c, false, false);
}

__global__ __launch_bounds__(256) void ssim_tile_kernel(
    const float* __restrict__ pred, const float* __restrict__ tgt,
    float* __restrict__ partial) {
    __shared__ float sp[HROWS * LP];
    __shared__ float st[HROWS * LP];
    __shared__ float sh[8 * 16 * HPITCH];   // per-wave H scratch, column-major
    __shared__ float red[8];

    const int tid   = threadIdx.x;
    const int wave  = tid >> 5;
    const int lane  = tid & 31;
    const int laneM = lane & 15;
    const int hi    = lane >> 4;

    const int bx   = blockIdx.x;
    const int img  = bx >> 7;          // / (TILES_X*TILES_Y) = /128
    const int rr_  = bx & 127;
    const int ty   = rr_ >> 3;
    const int tx   = rr_ & 7;
    const int gy0  = ty * 32;
    const int gx0  = tx * 64;
    const long base = (long)img * PLANE;

    // ---- cooperative zero-padded halo load (coalesced along x) ----
    for (int i = tid; i < HROWS * HCOLS; i += 256) {
        int r = i / HCOLS, c = i % HCOLS;
        int gr = gy0 - 5 + r, gc = gx0 - 5 + c;
        float pv = 0.f, tv = 0.f;
        if (gr >= 0 && gr < IMG_H && gc >= 0 && gc < IMG_W) {
            long idx = base + (long)gr * IMG_W + gc;
            pv = pred[idx];
            tv = tgt[idx];
        }
        sp[r * LP + c] = pv;
        st[r * LP + c] = tv;
    }
    __syncthreads();

    // ---- banded Gaussian fragments: serve as B for H-pass and A for V-pass
    //      element(k,v) = w[(4k + 2*hi + v) - laneM], zero outside [0,10]
    v2f wbv[7];
#pragma unroll
    for (int k = 0; k < 7; ++k) {
        int j0 = 4 * k + 2 * hi - laneM;
        int j1 = j0 + 1;
        wbv[k].x = (j0 >= 0 && j0 < 11) ? GW[j0] : 0.f;
        wbv[k].y = (j1 >= 0 && j1 < 11) ? GW[j1] : 0.f;
    }

    const int twy = wave >> 2, twx = wave & 3;   // 2x4 wave tiles of 16x16
    const int ly0 = 16 * twy, lx0 = 16 * twx;
    float* hb = &sh[wave * (16 * HPITCH)];

    // zero K-pad rows 26/27 of this lane's H column (weight there is 0)
    if (hi == 0) {
        v2f z; z.x = 0.f; z.y = 0.f;
        *(v2f*)&hb[laneM * HPITCH + 26] = z;
    }

    // ---- cache all pred/target A-fragments for both row blocks in VGPRs ----
    // (shared across the 5 conv inputs; 2rb x 7chunks x 2 tensors = 28 v2f)
    v2f pf[2][7], tf[2][7];
#pragma unroll
    for (int rb = 0; rb < 2; ++rb) {
        const int row = ly0 + 10 * rb + laneM;
        const float* prow = &sp[row * LP + lx0];
        const float* trow = &st[row * LP + lx0];
#pragma unroll
        for (int k = 0; k < 7; ++k) {
            int cb = 4 * k + 2 * hi;
            int c0 = cb > 25 ? 25 : cb;          // clamp pad cols: weight is 0 there
            int c1 = (cb + 1) > 25 ? 25 : (cb + 1);
            pf[rb][k].x = prow[c0]; pf[rb][k].y = prow[c1];
            tf[rb][k].x = trow[c0]; tf[rb][k].y = trow[c1];
        }
    }

    v8f res[5];
#pragma unroll
    for (int q = 0; q < 5; ++q) {
        // ---- horizontal pass: H(26x16) = In(26x26) x Wband, 2 row blocks ----
#pragma unroll
        for (int rb = 0; rb < 2; ++rb) {
            v8f acc = {};
#pragma unroll
            for (int k = 0; k < 7; ++k) {
                v2f p = pf[rb][k], t = tf[rb][k];
                v2f a;
                if (q == 0)      a = p;
                else if (q == 1) a = t;
                else if (q == 2) a = p * p;
                else if (q == 3) a = t * t;
                else             a = p * t;
                acc = wmma4(a, wbv[k], acc);
            }
            // C/D lane holds 8 consecutive rows of one H column -> b64 stores
            // (block rb covers H rows 10*rb+8*hi .. 10*rb+8*hi+7 of col laneM)
            float* cdst = &hb[laneM * HPITCH + 10 * rb + 8 * hi];
#pragma unroll
            for (int j = 0; j < 4; ++j) {
                v2f pr; pr.x = acc[2 * j]; pr.y = acc[2 * j + 1];
                *(v2f*)&cdst[2 * j] = pr;
            }
        }

        // Same-wave DS ops execute in order: these loads observe the stores
        // above; B fragment rows (4k+2*hi, +1) are contiguous in a column.
        v2f bf[7];
        const float* csrc = &hb[laneM * HPITCH + 2 * hi];
#pragma unroll
        for (int k = 0; k < 7; ++k)
            bf[k] = *(const v2f*)&csrc[4 * k];

        // ---- vertical pass: Out(16x16) = Wband(16x26) x H(26x16) ----
        v8f d = {};
#pragma unroll
        for (int k = 0; k < 7; ++k)
            d = wmma4(wbv[k], bf[k], d);
        res[q] = d;
    }

    // ---- elementwise SSIM map + in-register partial sum ----
    float lsum = 0.f;
#pragma unroll
    for (int v = 0; v < 8; ++v) {
        float m1 = res[0][v], m2 = res[1][v];
        float m1s = m1 * m1, m2s = m2 * m2, m12 = m1 * m2;
        float s11 = res[2][v] - m1s;
        float s22 = res[3][v] - m2s;
        float s12 = res[4][v] - m12;
        float num = (2.f * m12 + SSIM_C1) * (2.f * s12 + SSIM_C2);
        float den = (m1s + m2s + SSIM_C1) * (s11 + s22 + SSIM_C2);
        lsum += num / den;
    }

    // ---- wave32 reduce -> block reduce -> one deterministic partial per WG ----
#pragma unroll
    for (int off = 16; off > 0; off >>= 1)
        lsum += __shfl_down(lsum, off, 32);
    if (lane == 0) red[wave] = lsum;
    __syncthreads();
    if (tid == 0) {
        float s = 0.f;
#pragma unroll
        for (int w = 0; w < 8; ++w) s += red[w];
        partial[bx] = s;
    }
}

__global__ __launch_bounds__(256) void ssim_finalize_kernel(
    const float* __restrict__ partial, float* __restrict__ out) {
    __shared__ float red[256];
    const int tid = threadIdx.x;
    float s = 0.f;
    for (int i = tid; i < NWG; i += 256) s += partial[i];   // fixed order: deterministic
    red[tid] = s;
    __syncthreads();
    for (int off = 128; off > 0; off >>= 1) {
        if (tid < off) red[tid] += red[tid + off];
        __syncthreads();
    }
    if (tid == 0) out[0] = 1.f - red[0] / NTOT;
}

extern "C" void kernel_launch(void* const* d_in, const int* in_sizes, int n_in,
                              void* d_out, int out_size, void* d_ws, size_t ws_size,
                              hipStream_t stream) {
    (void)in_sizes; (void)n_in; (void)out_size; (void)ws_size;
    const float* pred = (const float*)d_in[0];
    const float* tgt  = (const float*)d_in[1];
    float* partial    = (float*)d_ws;      // NWG * 4 bytes
    float* out        = (float*)d_out;

    ssim_tile_kernel<<<dim3(NWG), dim3(256), 0, stream>>>(pred, tgt, partial);
    ssim_finalize_kernel<<<dim3(1), dim3(256), 0, stream>>>(partial, out);
}